// ClusterMapper_65712999628830
// MI455X (gfx1250) — compile-verified
//
#include <hip/hip_runtime.h>
#include <stdint.h>

#define CM_EPS 1e-6f

// Device-scope fp32 atomic add, no return (STOREcnt path, resolved in L2).
// Inline asm guarantees the native global_atomic_add_f32 (no CAS loop) and
// SCOPE_DEV coherence across WGPs.
__device__ __forceinline__ void atomic_add_f32_dev(float* p, float v) {
    asm volatile("global_atomic_add_f32 %0, %1, off scope:SCOPE_DEV"
                 :: "v"(p), "v"(v)
                 : "memory");
}

// Async copy of one DWORD per lane from global memory into LDS (ASYNCcnt path).
// lds_off is the byte offset of the destination inside the workgroup's LDS
// (low 32 bits of the generic pointer to a __shared__ object).
__device__ __forceinline__ void async_load_b32_to_lds(uint32_t lds_off, const void* gaddr) {
    asm volatile("global_load_async_to_lds_b32 %0, %1, off"
                 :: "v"(lds_off), "v"(gaddr)
                 : "memory");
}

__device__ __forceinline__ void wait_asynccnt0() {
    asm volatile("s_wait_asynccnt 0" ::: "memory");
}

__device__ __forceinline__ uint32_t lds_offset_of(const void* shared_generic_ptr) {
    // LDS aperture: generic address low 32 bits == LDS byte offset.
    return (uint32_t)(uintptr_t)shared_generic_ptr;
}

__global__ void cm_zero_f4(float4* __restrict__ p, long n4) {
    long i = (long)blockIdx.x * blockDim.x + threadIdx.x;
    if (i < n4) p[i] = make_float4(0.f, 0.f, 0.f, 0.f);
}

__global__ void cm_zero_f1(float* __restrict__ p, int n) {
    int i = blockIdx.x * blockDim.x + threadIdx.x;
    if (i < n) p[i] = 0.f;
}

// denom[m] += rd[n]  for m = mapping[n]
__global__ void cm_denom(const int* __restrict__ mapping,
                         const float* __restrict__ rd,
                         float* __restrict__ denom, int N) {
    int n = blockIdx.x * blockDim.x + threadIdx.x;
    if (n < N) {
        atomic_add_f32_dev(&denom[mapping[n]], rd[n]);
    }
}

// acc[bc*T + mapping[n]] += output[bc*N + n] * rd[n]
// Block: 256 threads handle 256 consecutive n; grid.y splits the BC channel
// rows so each thread amortizes one LDS-staged (mapping, rd) pair over
// bcPerY channel rows. output[] is streamed with non-temporal hints so it
// does not evict the L2-resident accumulator.
__global__ void cm_scatter(const float* __restrict__ outp,
                           const int* __restrict__ mapping,
                           const float* __restrict__ rd,
                           float* __restrict__ acc,
                           int N, int T, int BC, int bcPerY) {
    __shared__ int   smap[256];
    __shared__ float srd[256];

    const int tid = threadIdx.x;
    const int n   = blockIdx.x * 256 + tid;

    if (n < N) {
        async_load_b32_to_lds(lds_offset_of(&smap[tid]), (const void*)(mapping + n));
        async_load_b32_to_lds(lds_offset_of(&srd[tid]),  (const void*)(rd + n));
    }
    wait_asynccnt0();
    __syncthreads();

    if (n >= N) return;

    const int   m = smap[tid];
    const float w = srd[tid];

    const int bc0 = blockIdx.y * bcPerY;
    const float* __restrict__ src = outp + (long)bc0 * N + n;
    float*                    dst = acc  + (long)bc0 * T + m;

    const int iters = (bc0 + bcPerY <= BC) ? bcPerY : (BC - bc0);
    for (int i = 0; i < iters; ++i) {
        if (i + 1 < iters) __builtin_prefetch(src + N, 0, 0);
        float v = __builtin_nontemporal_load(src);   // streaming read, NT hint
        atomic_add_f32_dev(dst, v * w);              // L2-resident RMW
        src += N;
        dst += T;
    }
}

// out[row*T + t] /= max(denom[t], EPS), vectorized 4-wide.
// 2D launch: blockIdx.y = channel row (BC of them), columns cover T/4 float4s.
// No 64-bit modulo anywhere; denom (256 KB) stays L2-resident and is
// broadcast-read by every row.
__global__ void cm_divide4(float4* __restrict__ acc,
                           const float4* __restrict__ denom,
                           int T4) {
    int col = blockIdx.x * blockDim.x + threadIdx.x;
    if (col >= T4) return;
    long idx = (long)blockIdx.y * T4 + col;
    float4 a = acc[idx];
    float4 d = denom[col];
    a.x /= fmaxf(d.x, CM_EPS);
    a.y /= fmaxf(d.y, CM_EPS);
    a.z /= fmaxf(d.z, CM_EPS);
    a.w /= fmaxf(d.w, CM_EPS);
    acc[idx] = a;
}

// Scalar fallback for the divide (general T), also modulo-free via 2D grid.
__global__ void cm_divide1(float* __restrict__ acc,
                           const float* __restrict__ denom,
                           int T) {
    int col = blockIdx.x * blockDim.x + threadIdx.x;
    if (col >= T) return;
    long idx = (long)blockIdx.y * T + col;
    acc[idx] = acc[idx] / fmaxf(denom[col], CM_EPS);
}

extern "C" void kernel_launch(void* const* d_in, const int* in_sizes, int n_in,
                              void* d_out, int out_size, void* d_ws, size_t ws_size,
                              hipStream_t stream) {
    const float* outp    = (const float*)d_in[0];  // (B,C,H,W) float32
    const int*   mapping = (const int*)  d_in[1];  // (N,) int32
    const float* rd      = (const float*)d_in[2];  // (N,) float32
    // d_in[3] = n_targets (device scalar); T derived on host instead.

    float* out   = (float*)d_out;
    float* denom = (float*)d_ws;                   // T floats of scratch

    const long totalIn = in_sizes[0];              // B*C*N
    const int  N       = in_sizes[1];              // H*W
    const int  BC      = (int)(totalIn / N);       // B*C
    const int  T       = out_size / BC;            // n_targets

    const int BS = 256;

    // 1) zero accumulator (d_out) and denominator scratch
    {
        long out4 = (long)out_size / 4;            // out_size = BC*T with BC=256 -> %4==0
        cm_zero_f4<<<dim3((unsigned)((out4 + BS - 1) / BS)), BS, 0, stream>>>(
            (float4*)out, out4);
        cm_zero_f1<<<dim3((unsigned)((T + BS - 1) / BS)), BS, 0, stream>>>(denom, T);
    }

    // 2) denominator scatter-add
    cm_denom<<<dim3((unsigned)((N + BS - 1) / BS)), BS, 0, stream>>>(mapping, rd, denom, N);

    // 3) main weighted scatter-add, channel-split 8 ways
    {
        const int YS = 8;
        const int bcPerY = (BC + YS - 1) / YS;
        dim3 grid((unsigned)((N + BS - 1) / BS), (unsigned)YS);
        cm_scatter<<<grid, BS, 0, stream>>>(outp, mapping, rd, out, N, T, BC, bcPerY);
    }

    // 4) normalize (modulo-free 2D launch)
    {
        if ((T & 3) == 0) {
            int T4 = T / 4;
            dim3 grid((unsigned)((T4 + BS - 1) / BS), (unsigned)BC);
            cm_divide4<<<grid, BS, 0, stream>>>((float4*)out, (const float4*)denom, T4);
        } else {
            dim3 grid((unsigned)((T + BS - 1) / BS), (unsigned)BC);
            cm_divide1<<<grid, BS, 0, stream>>>(out, denom, T);
        }
    }
}